// HBVMulTDET_water_loss_472446402986
// MI455X (gfx1250) — compile-verified
//
#include <hip/hip_runtime.h>
#include <hip/hip_bf16.h>

typedef __attribute__((ext_vector_type(2))) float v2f;
typedef __attribute__((ext_vector_type(8))) float v8f;

#define PRECS_F 1e-5f

__device__ __forceinline__ float fclampf(float x, float lo, float hi) {
    return fminf(fmaxf(x, lo), hi);
}

// ---------------------------------------------------------------------------
// Kernel 1: HBV sequential scan. One thread per (grid g, ensemble member m).
// mu=16 members occupy 16 consecutive lanes -> mean over mu via shfl_xor.
// Writes q scaled by Ai/mu, TRANSPOSED: qT[g * Tpad + t] (Tpad = 736) so the
// downstream WMMA GEMM reads A coalesced along t.
// ---------------------------------------------------------------------------
__global__ void __launch_bounds__(256)
hbv_scan_kernel(const float* __restrict__ x,          // (Nstep, Ngrid, 3)
                const float* __restrict__ params,     // (Nstep, Ngrid, 3, mu)
                const float* __restrict__ wl,         // (Ngrid, 13, mu)
                const float* __restrict__ Ai_b,       // (Ngrid)
                const float* __restrict__ Ac_b,       // (Ngrid)
                float* __restrict__ qT,               // (Ngrid, Tpad) workspace
                int Nstep, int Ngrid, int mu, int Tpad)
{
    const int tid = blockIdx.x * blockDim.x + threadIdx.x;
    const int m = tid % mu;
    const int g = tid / mu;
    if (g >= Ngrid) return;

    // ---- per-(g,m) waterloss parameters: lo + u * (hi - lo), SCA2 rows ----
    const float* w = wl + (size_t)g * 13 * mu + m;
    const float parFC    = 50.0f  + w[0 * mu]  * 950.0f;
    const float parK1    = 0.01f  + w[1 * mu]  * 0.49f;
    const float parK2    = 0.001f + w[2 * mu]  * 0.199f;
    const float parLP    = 0.2f   + w[3 * mu]  * 0.8f;
    const float parPERC  =          w[4 * mu]  * 10.0f;
    const float parUZL   =          w[5 * mu]  * 100.0f;
    const float parTT    = -2.5f  + w[6 * mu]  * 5.0f;
    const float parCFMAX = 0.5f   + w[7 * mu]  * 9.5f;
    const float parCFR   =          w[8 * mu]  * 0.1f;
    const float parCWH   =          w[9 * mu]  * 0.2f;
    const float parC     =          w[10 * mu] * 1.0f;
    const float parTR    =          w[11 * mu] * 20.0f;
    const float parAc    =          w[12 * mu] * 2500.0f;

    const float Ac = Ac_b[g];
    float rf;
    if (Ac < 2500.0f)
        rf = fclampf((Ac - parAc) * 0.001f, -1.0f, 1.0f) * parTR;
    else
        rf = __expf(fclampf(-(Ac - 2500.0f) * 0.02f, -10.0f, 0.0f)) * parTR;

    const float Aig = Ai_b[g] / (float)mu;
    const float invFC = 1.0f / parFC;
    const float invLPFC = 1.0f / (parLP * parFC);

    float SNOWPACK = 0.001f, MELTWATER = 0.001f, SM = 0.001f;
    float SUZ = 0.001f, SLZ = 0.001f;

    for (int t = 0; t < Nstep; ++t) {
        const float* xp = x + ((size_t)t * Ngrid + g) * 3;
        const float Pt = xp[0], Tt = xp[1], Et = xp[2];

        const float* pp = params + (((size_t)t * Ngrid + g) * 3) * mu + m;
        const float parBETA   = 1.0f  + pp[0]      * 5.0f;   // SCA1 row 0
        const float parK0     = 0.05f + pp[mu]     * 0.85f;  // SCA1 row 1
        const float parBETAET = 0.3f  + pp[2 * mu] * 4.7f;   // SCA1 row 2

        const float RAIN = (Tt >= parTT) ? Pt : 0.0f;
        const float SNOW = Pt - RAIN;
        SNOWPACK += SNOW;
        const float melt = fminf(fmaxf(parCFMAX * (Tt - parTT), 0.0f), SNOWPACK);
        MELTWATER += melt;
        SNOWPACK  -= melt;
        const float refr = fminf(fmaxf(parCFR * parCFMAX * (parTT - Tt), 0.0f), MELTWATER);
        SNOWPACK  += refr;
        MELTWATER -= refr;
        const float tosoil = fmaxf(MELTWATER - parCWH * SNOWPACK, 0.0f);
        MELTWATER -= tosoil;

        const float sw = fclampf(__powf(SM * invFC, parBETA), 0.0f, 1.0f);
        const float recharge = (RAIN + tosoil) * sw;
        SM += RAIN + tosoil - recharge;
        const float excess = fmaxf(SM - parFC, 0.0f);
        SM -= excess;
        const float ef = fclampf(__powf(SM * invLPFC, parBETAET), 0.0f, 1.0f);
        const float ETact = fminf(SM, Et * ef);
        SM = fmaxf(SM - ETact, PRECS_F);
        const float cap = fminf(SLZ, parC * SLZ * (1.0f - fminf(SM * invFC, 1.0f)));
        SM  = fmaxf(SM + cap, PRECS_F);
        SLZ = fmaxf(SLZ - cap, PRECS_F);

        SUZ += recharge + excess;
        const float PERC = fminf(SUZ, parPERC);
        SUZ -= PERC;
        const float Q0 = parK0 * fmaxf(SUZ - parUZL, 0.0f);
        SUZ -= Q0;
        const float Q1 = parK1 * SUZ;
        SUZ -= Q1;
        SLZ += PERC;
        SLZ = fmaxf(SLZ + rf, 0.0f);
        const float Q2 = parK2 * SLZ;
        SLZ -= Q2;

        float q = Q0 + Q1 + Q2;
        // mean over mu=16 lanes (lane groups of mu within the wave)
        for (int off = mu >> 1; off > 0; off >>= 1)
            q += __shfl_xor(q, off, mu);
        if (m == 0)
            qT[(size_t)g * Tpad + t] = q * Aig;
    }
}

// ---------------------------------------------------------------------------
// Kernel 2: Qsim[t][n] = sum_g qT[g][t] * idx[g][n]   via V_WMMA_F32_16X16X4_F32
// One wave per 16x16 output tile; K = Ngrid in steps of 4.
// A layout: lane = M (0..15 per half-wave); v.x -> K = k0 + 2*(lane>=16),
//           v.y -> K+1.   B mirrored with lane = N.
// C/D layout: VGPR r -> M = r + 8*(lane>=16), N = lane%16.
// ---------------------------------------------------------------------------
__global__ void __launch_bounds__(256)
qagg_wmma_kernel(const float* __restrict__ qT,    // (Ngrid, Tpad)
                 const float* __restrict__ idxm,  // (Ngrid, Nout)
                 float* __restrict__ out,         // (Nstep, Nout)
                 int Nstep, int Ngrid, int Nout, int Tpad,
                 int tiles_n, int total_tiles)
{
    const int wave = (int)((blockIdx.x * blockDim.x + threadIdx.x) >> 5);
    const int lane = threadIdx.x & 31;
    if (wave >= total_tiles) return;   // wave-uniform

    const int tile_m = wave / tiles_n;
    const int tile_n = wave % tiles_n;
    const int t0 = tile_m * 16;
    const int n0 = tile_n * 16;

    const int lm = lane & 15;            // M for A-frag / N for B-frag
    const int kh = (lane >> 4) << 1;     // K offset within 4-block per half-wave
    const int ncol = n0 + lm;
    const int nclamp = min(ncol, Nout - 1);   // tail cols land in unstored lanes

    v8f c = {};
    for (int k0 = 0; k0 < Ngrid; k0 += 4) {
        const int ka = k0 + kh;
        v2f a, b;
        a.x = qT[(size_t)ka * Tpad + t0 + lm];
        a.y = qT[(size_t)(ka + 1) * Tpad + t0 + lm];
        b.x = idxm[(size_t)ka * Nout + nclamp];
        b.y = idxm[(size_t)(ka + 1) * Nout + nclamp];
        c = __builtin_amdgcn_wmma_f32_16x16x4_f32(
                /*neg_a=*/false, a, /*neg_b=*/false, b,
                /*c_mod=*/(short)0, c, /*reuse_a=*/false, /*reuse_b=*/false);
    }

    if (ncol < Nout) {
        const int mbase = (lane >> 4) << 3;
#pragma unroll
        for (int r = 0; r < 8; ++r) {
            const int row = t0 + mbase + r;
            if (row < Nstep)
                out[(size_t)row * Nout + ncol] = c[r];
        }
    }
}

// ---------------------------------------------------------------------------
extern "C" void kernel_launch(void* const* d_in, const int* in_sizes, int n_in,
                              void* d_out, int out_size, void* d_ws, size_t ws_size,
                              hipStream_t stream) {
    const float* x      = (const float*)d_in[0];  // (Nstep, Ngrid, 3)
    const float* params = (const float*)d_in[1];  // (Nstep, Ngrid, 3, mu)
    const float* wl     = (const float*)d_in[2];  // (Ngrid, 13, mu)
    const float* Ai_b   = (const float*)d_in[3];  // (Ngrid)
    const float* Ac_b   = (const float*)d_in[4];  // (Ngrid)
    const float* idxm   = (const float*)d_in[5];  // (Ngrid, Nout)
    float* out = (float*)d_out;

    const int Ngrid = in_sizes[3];
    const int Nstep = in_sizes[0] / (Ngrid * 3);
    const int mu    = in_sizes[1] / (Nstep * Ngrid * 3);
    const int Nout  = in_sizes[5] / Ngrid;
    const int Tpad  = ((Nstep + 15) / 16) * 16;   // 730 -> 736

    float* qT = (float*)d_ws;                     // needs Ngrid*Tpad*4 bytes (~5.9 MB)

    // Kernel 1: scan over time, parallel over (grid, mu)
    {
        const int threads = Ngrid * mu;           // 32000
        const int block = 256;
        const int grid = (threads + block - 1) / block;
        hbv_scan_kernel<<<grid, block, 0, stream>>>(
            x, params, wl, Ai_b, Ac_b, qT, Nstep, Ngrid, mu, Tpad);
    }

    // Kernel 2: (Nstep x Ngrid) @ (Ngrid x Nout) fp32 WMMA GEMM
    {
        const int tiles_m = (Nstep + 15) / 16;    // 46
        const int tiles_n = (Nout + 15) / 16;     // 32
        const int total_tiles = tiles_m * tiles_n;
        const int block = 256;                    // 8 waves per block
        const int grid = (total_tiles * 32 + block - 1) / block;
        qagg_wmma_kernel<<<grid, block, 0, stream>>>(
            qT, idxm, out, Nstep, Ngrid, Nout, Tpad, tiles_n, total_tiles);
    }
}